// ParallelIFS_39462159516154
// MI455X (gfx1250) — compile-verified
//
#include <hip/hip_runtime.h>
#include <stdint.h>

#define MB     2048
#define NP     8192
#define NF     8
#define RM     51
#define NCHUNK 64
#define CHUNK  128            // NP / NCHUNK
#define TSTEP  32             // t-steps per async tile
#define NTILE  (CHUNK / TSTEP)
#define ROWSTRIDE 36          // ints per staged row (32 + 4 pad) = 144B, 16B aligned

// ---------------- Kernel 1: per-(batch,chunk) affine composition ----------------
__global__ void __launch_bounds__(256) ifs_compose(
    const float* __restrict__ w, const float* __restrict__ bias,
    const int* __restrict__ index, float* __restrict__ trans) {
  __shared__ float4 tab[NF * 2];
  int tid = threadIdx.x;
  if (tid < NF) {
    tab[tid * 2]     = make_float4(w[tid*4+0], w[tid*4+1], w[tid*4+2], w[tid*4+3]);
    tab[tid * 2 + 1] = make_float4(bias[tid*2+0], bias[tid*2+1], 0.f, 0.f);
  }
  __syncthreads();
  int b = blockIdx.x * 256 + tid;
  int k = blockIdx.y;
  const int4* ip = reinterpret_cast<const int4*>(index + (size_t)b * NP + k * CHUNK);
  float a00 = 1.f, a01 = 0.f, a10 = 0.f, a11 = 1.f, c0 = 0.f, c1 = 0.f;
#pragma unroll 4
  for (int q = 0; q < CHUNK / 4; ++q) {
    int4 iv = ip[q];
    int fs[4] = {iv.x, iv.y, iv.z, iv.w};
#pragma unroll
    for (int u = 0; u < 4; ++u) {
      float4 W = tab[fs[u] * 2];
      float4 B = tab[fs[u] * 2 + 1];
      float na00 = W.x*a00 + W.y*a10, na01 = W.x*a01 + W.y*a11;
      float na10 = W.z*a00 + W.w*a10, na11 = W.z*a01 + W.w*a11;
      float nc0  = B.x + W.x*c0 + W.y*c1;
      float nc1  = B.y + W.z*c0 + W.w*c1;
      a00 = na00; a01 = na01; a10 = na10; a11 = na11; c0 = nc0; c1 = nc1;
    }
  }
  const int P = NCHUNK * MB;
  int o = k * MB + b;
  trans[0*P+o] = a00; trans[1*P+o] = a01; trans[2*P+o] = a10;
  trans[3*P+o] = a11; trans[4*P+o] = c0;  trans[5*P+o] = c1;
}

// ---------------- Kernel 2: per-batch scan over chunk transforms ----------------
__global__ void __launch_bounds__(256) ifs_scan(
    const float* __restrict__ point, const float* __restrict__ trans,
    float* __restrict__ start) {
  int b = blockIdx.x * 256 + threadIdx.x;
  float px = point[b * 2 + 0], py = point[b * 2 + 1];
  const int P = NCHUNK * MB;
  for (int k = 0; k < NCHUNK; ++k) {
    int o = k * MB + b;
    start[0*P + o] = px;
    start[1*P + o] = py;
    float a00 = trans[0*P+o], a01 = trans[1*P+o], a10 = trans[2*P+o];
    float a11 = trans[3*P+o], c0  = trans[4*P+o], c1  = trans[5*P+o];
    float nx = c0 + a00 * px + a01 * py;
    float ny = c1 + a10 * px + a11 * py;
    px = nx; py = ny;
  }
}

// ---------------- Kernel 3: replay + streaming stores, async index staging ------
__global__ void __launch_bounds__(32) ifs_replay(
    const float* __restrict__ w, const float* __restrict__ bias,
    const float* __restrict__ ops, const int* __restrict__ index,
    const float* __restrict__ start, float* __restrict__ out) {
  __shared__ float4 tab[NF * 2];
  __shared__ __align__(16) int stage[2][32 * ROWSTRIDE];
  int lane = threadIdx.x;
  if (lane < NF) {
    tab[lane * 2]     = make_float4(w[lane*4+0], w[lane*4+1], w[lane*4+2], w[lane*4+3]);
    tab[lane * 2 + 1] = make_float4(bias[lane*2+0], bias[lane*2+1], ops[lane], 0.f);
  }
  __syncthreads();

  int bBase = blockIdx.x * 32;
  int b     = bBase + lane;
  int k     = blockIdx.y;
  const int P = NCHUNK * MB;
  float px = start[0*P + k*MB + b];
  float py = start[1*P + k*MB + b];

  // Async stage: SGPR base + per-lane u32 byte offset (GVS). One b128 op moves
  // 512B (4 index rows x 128B); 8 ops cover the wave's 32 rows x 32 steps.
  const int* sbase = index + (size_t)bBase * NP + k * CHUNK;
  unsigned row   = lane >> 3;
  unsigned seg   = (lane & 7u) * 16u;
  unsigned vg0   = row * (NP * 4u) + seg;              // global byte offset, op j adds j*4*NP*4
  unsigned ldsr  = row * (ROWSTRIDE * 4u) + seg;
  unsigned lds0  = (unsigned)(uintptr_t)(&stage[0][0]);
  unsigned lds1  = (unsigned)(uintptr_t)(&stage[1][0]);

  auto issue_tile = [&](int tile, int bufsel) {
    unsigned ldsbase = (bufsel ? lds1 : lds0) + ldsr;
    unsigned gbase   = vg0 + (unsigned)tile * (TSTEP * 4u);
#pragma unroll
    for (int j = 0; j < 8; ++j) {
      unsigned ldsa = ldsbase + (unsigned)j * (4u * ROWSTRIDE * 4u);
      unsigned ga   = gbase   + (unsigned)j * (4u * NP * 4u);
      asm volatile("global_load_async_to_lds_b128 %0, %1, %2"
                   :: "v"(ldsa), "v"(ga), "s"(sbase) : "memory");
    }
  };

  issue_tile(0, 0);
  int buf = 0;
  for (int tile = 0; tile < NTILE; ++tile) {
    if (tile + 1 < NTILE) issue_tile(tile + 1, buf ^ 1);
#if __has_builtin(__builtin_amdgcn_s_wait_asynccnt)
    if (tile + 1 < NTILE) __builtin_amdgcn_s_wait_asynccnt(8);
    else                  __builtin_amdgcn_s_wait_asynccnt(0);
#else
    if (tile + 1 < NTILE) asm volatile("s_wait_asynccnt 0x8" ::: "memory");
    else                  asm volatile("s_wait_asynccnt 0x0" ::: "memory");
#endif
    asm volatile("" ::: "memory");

    const int* myrow = &stage[buf][lane * ROWSTRIDE];
    int t0 = k * CHUNK + tile * TSTEP;
#pragma unroll
    for (int q = 0; q < TSTEP / 4; ++q) {
      int4 iv = *reinterpret_cast<const int4*>(myrow + q * 4);
      int fs[4] = {iv.x, iv.y, iv.z, iv.w};
#pragma unroll
      for (int u = 0; u < 4; ++u) {
        int t = t0 + q * 4 + u;
        float4 W = tab[fs[u] * 2];
        float4 B = tab[fs[u] * 2 + 1];
        float nx = B.x + W.x * px + W.y * py;
        float ny = B.y + W.z * px + W.w * py;
        px = nx; py = ny;
        if (t >= RM) {
          size_t r = (size_t)(t - RM) * MB + b;   // wave writes 384B contiguous
          out[r * 3 + 0] = px;
          out[r * 3 + 1] = py;
          out[r * 3 + 2] = B.z;
        }
      }
    }
    buf ^= 1;
  }
}

extern "C" void kernel_launch(void* const* d_in, const int* in_sizes, int n_in,
                              void* d_out, int out_size, void* d_ws, size_t ws_size,
                              hipStream_t stream) {
  const float* point = (const float*)d_in[0];   // [2048,2,1]
  const float* w     = (const float*)d_in[1];   // [8,2,2]
  const float* bias  = (const float*)d_in[2];   // [8,2,1]
  const float* ops   = (const float*)d_in[3];   // [8]
  const int*   index = (const int*)d_in[4];     // [2048,8192]
  float* out = (float*)d_out;

  // ws: 6 planes of chunk transforms + 2 planes of start points = 4 MB
  float* trans = (float*)d_ws;
  float* start = trans + 6 * NCHUNK * MB;

  ifs_compose<<<dim3(MB / 256, NCHUNK), 256, 0, stream>>>(w, bias, index, trans);
  ifs_scan<<<dim3(MB / 256), 256, 0, stream>>>(point, trans, start);
  ifs_replay<<<dim3(MB / 32, NCHUNK), 32, 0, stream>>>(w, bias, ops, index, start, out);
}